// GATModel_1039382086072
// MI455X (gfx1250) — compile-verified
//
#include <hip/hip_runtime.h>
#include <math.h>

#define NNODES 100000
#define NEDGES 400000
#define NGRAPH 4096
#define DIN    74
#define DHID   200
#define NLAYER 5
#define DMLP   1024
#define SLOPE  0.2f

typedef __attribute__((ext_vector_type(2))) float v2f;
typedef __attribute__((ext_vector_type(8))) float v8f;

// gfx1250 async global->LDS staging (ASYNCcnt path), if the toolchain has it.
#if defined(__has_builtin)
# if __has_builtin(__builtin_amdgcn_global_load_async_to_lds_b32) && \
     __has_builtin(__builtin_amdgcn_s_wait_asynccnt)
#  define USE_ASYNC_LDS 1
# endif
#endif
#ifndef USE_ASYNC_LDS
# define USE_ASYNC_LDS 0
#endif

// Builtin prototype (per clang diagnostic): (global int*, shared int*, imm, imm)
#define AS1I(p) ((__attribute__((address_space(1))) int*)(p))
#define AS3I(p) ((__attribute__((address_space(3))) int*)(p))

// ---------------------------------------------------------------------------
// Generic fill
// ---------------------------------------------------------------------------
__global__ void fill_kernel(float* __restrict__ p, int n, float v) {
  int i = blockIdx.x * blockDim.x + threadIdx.x;
  if (i < n) p[i] = v;
}

// ---------------------------------------------------------------------------
// WMMA fp32 GEMM: C[M,N] = A[M,K] @ B[K,N] (+ bias[col]) (+ optional ReLU)
//
// Block = 128 threads (4 waves) -> one 32x32 output tile (2x2 wave tiles).
// K consumed in chunks of 32: stage A (row-major) and B (transposed) into
// LDS, then each wave runs 8 unguarded V_WMMA_F32_16X16X4_F32 steps fed by
// ds_load_b64 fragment reads. Fragment layout per ISA (32-bit A 16x4/B 4x16):
// lanes 0-15 hold K={k,k+1}, lanes 16-31 hold K={k+2,k+3}; C/D VGPR v ->
// rows v (lanes 0-15) and v+8 (lanes 16-31).
// ---------------------------------------------------------------------------
#define TK  32
#define LDP 36   // LDS pitch in floats (144 B: keeps rows 16B-aligned, skews banks)

__global__ void __launch_bounds__(128)
wmma_gemm_f32(const float* __restrict__ A,
              const float* __restrict__ B,
              const float* __restrict__ bias,
              float* __restrict__ C,
              int M, int K, int N, int relu) {
  __shared__ float ldsA[32 * LDP];    // A tile: [r][k]
  __shared__ float ldsBt[32 * LDP];   // B tile transposed: [c][k]

  const int tid  = threadIdx.x;
  const int lane = tid & 31;
  const int w    = tid >> 5;          // wave 0..3
  const int wm   = (w >> 1) << 4;     // 0 / 16: wave tile row offset
  const int wn   = (w & 1) << 4;      // 0 / 16: wave tile col offset
  const int half = lane >> 4;         // 0: lanes 0-15, 1: lanes 16-31
  const int lr   = lane & 15;

  const int tilesN = (N + 31) >> 5;
  const int tM2 = (blockIdx.x / tilesN) << 5;
  const int tN2 = (blockIdx.x % tilesN) << 5;

  const bool interiorMN = (tM2 + 32 <= M) && (tN2 + 32 <= N);

  v8f acc = {};
  for (int k0 = 0; k0 < K; k0 += TK) {
#if USE_ASYNC_LDS
    if (interiorMN && (k0 + TK <= K)) {
      // Fast path: async copy global->LDS, per-lane addresses (B transposes
      // in flight via the LDS destination address).
      for (int idx = tid; idx < 32 * TK; idx += 128) {
        const int r  = idx >> 5, ka = idx & 31;          // A: k fastest (coalesced)
        __builtin_amdgcn_global_load_async_to_lds_b32(
            AS1I(A + (size_t)(tM2 + r) * K + (k0 + ka)),
            AS3I(&ldsA[r * LDP + ka]), 0, 0);
        const int c  = idx & 31, kb = idx >> 5;          // B: c fastest (coalesced)
        __builtin_amdgcn_global_load_async_to_lds_b32(
            AS1I(B + (size_t)(k0 + kb) * N + (tN2 + c)),
            AS3I(&ldsBt[c * LDP + kb]), 0, 0);
      }
      __builtin_amdgcn_s_wait_asynccnt(0);
      __syncthreads();
    } else
#endif
    {
      // Guarded path (boundary tiles / K remainder): zero-padded staging.
      for (int idx = tid; idx < 32 * TK; idx += 128) {
        const int r = idx >> 5, ka = idx & 31;
        const int gr = tM2 + r, gka = k0 + ka;
        ldsA[r * LDP + ka] =
            (gr < M && gka < K) ? A[(size_t)gr * K + gka] : 0.0f;
        const int c = idx & 31, kb = idx >> 5;
        const int gc = tN2 + c, gkb = k0 + kb;
        ldsBt[c * LDP + kb] =
            (gkb < K && gc < N) ? B[(size_t)gkb * N + gc] : 0.0f;
      }
      __syncthreads();
    }

    // Branch-free WMMA chunk: 8 x (16x16x4) = K of 32.
    const float* pa = &ldsA[(wm + lr) * LDP + 2 * half];
    const float* pb = &ldsBt[(wn + lr) * LDP + 2 * half];
#pragma unroll
    for (int kk = 0; kk < 8; ++kk) {
      v2f a = *(const v2f*)(pa + kk * 4);   // ds_load_b64
      v2f b = *(const v2f*)(pb + kk * 4);   // ds_load_b64
      acc = __builtin_amdgcn_wmma_f32_16x16x4_f32(
          /*neg_a=*/false, a, /*neg_b=*/false, b,
          /*c_mod=*/(short)0, acc, /*reuse_a=*/false, /*reuse_b=*/false);
    }
    __syncthreads();
  }

  const int colB = tN2 + wn + lr;
  const bool cOK = (colB < N);
  const float bcol = (bias != nullptr && cOK) ? bias[colB] : 0.0f;
#pragma unroll
  for (int v = 0; v < 8; ++v) {
    const int row = tM2 + wm + v + 8 * half;
    if (row < M && cOK) {
      float val = acc[v] + bcol;
      if (relu) val = fmaxf(val, 0.0f);
      C[(size_t)row * N + colB] = val;
    }
  }
}

// ---------------------------------------------------------------------------
// el[n] = z[n,:].al ; er[n] = z[n,:].ar  (one wave per node, wave32 reduce)
// ---------------------------------------------------------------------------
__global__ void node_dots_kernel(const float* __restrict__ z,
                                 const float* __restrict__ al,
                                 const float* __restrict__ ar,
                                 float* __restrict__ el,
                                 float* __restrict__ er) {
  const int lane = threadIdx.x & 31;
  const int node = (int)((blockIdx.x * (size_t)blockDim.x + threadIdx.x) >> 5);
  if (node >= NNODES) return;
  const float* zr = z + (size_t)node * DHID;
  float sl = 0.0f, sr = 0.0f;
  for (int d = lane; d < DHID; d += 32) {
    float v = zr[d];
    sl += v * al[d];
    sr += v * ar[d];
  }
  for (int off = 16; off > 0; off >>= 1) {
    sl += __shfl_down(sl, off, 32);
    sr += __shfl_down(sr, off, 32);
  }
  if (lane == 0) { el[node] = sl; er[node] = sr; }
}

// ---------------------------------------------------------------------------
// Edge pass 1: e = leakyrelu(el[src]+er[dst]); m[dst] = max(m[dst], e)
// ---------------------------------------------------------------------------
__global__ void edge_score_max_kernel(const int* __restrict__ src,
                                      const int* __restrict__ dst,
                                      const float* __restrict__ el,
                                      const float* __restrict__ er,
                                      float* __restrict__ e,
                                      float* __restrict__ m) {
  int j = blockIdx.x * blockDim.x + threadIdx.x;
  if (j >= NEDGES) return;
  float v = el[src[j]] + er[dst[j]];
  v = (v > 0.0f) ? v : SLOPE * v;
  e[j] = v;
  atomicMax(&m[dst[j]], v);   // float max atomic (global_atomic_max_num_f32)
}

// ---------------------------------------------------------------------------
// Edge pass 2: ex = exp(e - m[dst]); denom[dst] += ex
// ---------------------------------------------------------------------------
__global__ void edge_exp_sum_kernel(const int* __restrict__ dst,
                                    const float* __restrict__ e,
                                    const float* __restrict__ m,
                                    float* __restrict__ ex,
                                    float* __restrict__ denom) {
  int j = blockIdx.x * blockDim.x + threadIdx.x;
  if (j >= NEDGES) return;
  int d = dst[j];
  float x = __expf(e[j] - m[d]);
  ex[j] = x;
  atomicAdd(&denom[d], x);
}

// ---------------------------------------------------------------------------
// Edge pass 3: agg[dst,:] += (ex/denom[dst]) * z[src,:]   (one wave per edge)
// ---------------------------------------------------------------------------
__global__ void edge_aggregate_kernel(const int* __restrict__ src,
                                      const int* __restrict__ dst,
                                      const float* __restrict__ ex,
                                      const float* __restrict__ denom,
                                      const float* __restrict__ z,
                                      float* __restrict__ agg) {
  const int lane = threadIdx.x & 31;
  const int j = (int)((blockIdx.x * (size_t)blockDim.x + threadIdx.x) >> 5);
  if (j >= NEDGES) return;
  const int s = src[j], t = dst[j];
  const float a = ex[j] / denom[t];
  const float* zs = z + (size_t)s * DHID;
  float* ag = agg + (size_t)t * DHID;
  for (int d = lane; d < DHID; d += 32)
    atomicAdd(&ag[d], a * zs[d]);
}

// ---------------------------------------------------------------------------
// h = act(agg + h + b[col]) ; ELU when activate != 0
// ---------------------------------------------------------------------------
__global__ void residual_act_kernel(const float* __restrict__ agg,
                                    const float* __restrict__ b,
                                    float* __restrict__ h,
                                    int activate) {
  int i = blockIdx.x * blockDim.x + threadIdx.x;
  if (i >= NNODES * DHID) return;
  float v = agg[i] + h[i] + b[i % DHID];
  if (activate) v = (v > 0.0f) ? v : (__expf(v) - 1.0f);
  h[i] = v;
}

// ---------------------------------------------------------------------------
// Per-graph readout: g[graph_ids[n], d] += h[n, d]
// ---------------------------------------------------------------------------
__global__ void readout_kernel(const float* __restrict__ h,
                               const int* __restrict__ gid,
                               float* __restrict__ g) {
  int i = blockIdx.x * blockDim.x + threadIdx.x;
  if (i >= NNODES * DHID) return;
  int n = i / DHID, d = i - n * DHID;
  atomicAdd(&g[(size_t)gid[n] * DHID + d], h[i]);
}

// ---------------------------------------------------------------------------
// out[g] = hidden[g,:] . W2 + b2   (one wave per graph)
// ---------------------------------------------------------------------------
__global__ void mlp_out_kernel(const float* __restrict__ hidden,
                               const float* __restrict__ W2,
                               const float* __restrict__ b2,
                               float* __restrict__ out) {
  const int lane = threadIdx.x & 31;
  const int gi = (int)((blockIdx.x * (size_t)blockDim.x + threadIdx.x) >> 5);
  if (gi >= NGRAPH) return;
  const float* hr = hidden + (size_t)gi * DMLP;
  float s = 0.0f;
  for (int k = lane; k < DMLP; k += 32) s += hr[k] * W2[k];
  for (int off = 16; off > 0; off >>= 1) s += __shfl_down(s, off, 32);
  if (lane == 0) out[gi] = s + b2[0];
}

// ---------------------------------------------------------------------------
extern "C" void kernel_launch(void* const* d_in, const int* in_sizes, int n_in,
                              void* d_out, int out_size, void* d_ws, size_t ws_size,
                              hipStream_t stream) {
  const float* node_feat = (const float*)d_in[0];
  const float* W0        = (const float*)d_in[1];
  const float* b0        = (const float*)d_in[2];
  const float* Ws        = (const float*)d_in[3];   // [L, D, D]
  const float* attn_l    = (const float*)d_in[4];   // [L, D]
  const float* attn_r    = (const float*)d_in[5];   // [L, D]
  const float* bs        = (const float*)d_in[6];   // [L, D]
  const float* W1        = (const float*)d_in[7];   // [D, 1024]
  const float* b1        = (const float*)d_in[8];
  const float* W2        = (const float*)d_in[9];   // [1024, 1]
  const float* b2        = (const float*)d_in[10];
  const int*   src       = (const int*)d_in[11];
  const int*   dst       = (const int*)d_in[12];
  const int*   gid       = (const int*)d_in[13];
  float*       out       = (float*)d_out;

  // Carve workspace
  char* ws = (char*)d_ws;
  size_t off = 0;
  auto carve = [&](size_t bytes) -> float* {
    float* p = (float*)(ws + off);
    off += (bytes + 255) & ~(size_t)255;
    return p;
  };
  const size_t ND = (size_t)NNODES * DHID;
  float* h      = carve(ND * 4);
  float* z      = carve(ND * 4);
  float* agg    = carve(ND * 4);
  float* el     = carve((size_t)NNODES * 4);
  float* er     = carve((size_t)NNODES * 4);
  float* m      = carve((size_t)NNODES * 4);
  float* denom  = carve((size_t)NNODES * 4);
  float* e      = carve((size_t)NEDGES * 4);
  float* ex     = carve((size_t)NEDGES * 4);
  float* g      = carve((size_t)NGRAPH * DHID * 4);
  float* hidden = carve((size_t)NGRAPH * DMLP * 4);

  auto gemm_grid = [](int M, int N) {
    return ((M + 31) >> 5) * ((N + 31) >> 5);   // one 32x32 tile per block
  };

  // Encoder: h = node_feat @ W0 + b0
  wmma_gemm_f32<<<gemm_grid(NNODES, DHID), 128, 0, stream>>>(
      node_feat, W0, b0, h, NNODES, DIN, DHID, 0);

  const int nodeWaveBlocks = (NNODES * 32 + 255) / 256;
  const int edgeBlocks     = (NEDGES + 255) / 256;
  const int edgeWaveBlocks = (NEDGES * 32 + 255) / 256;
  const int ndBlocks       = (int)((ND + 255) / 256);

  for (int l = 0; l < NLAYER; ++l) {
    const float* Wl  = Ws     + (size_t)l * DHID * DHID;
    const float* all = attn_l + (size_t)l * DHID;
    const float* arl = attn_r + (size_t)l * DHID;
    const float* bl  = bs     + (size_t)l * DHID;

    // z = h @ W
    wmma_gemm_f32<<<gemm_grid(NNODES, DHID), 128, 0, stream>>>(
        h, Wl, nullptr, z, NNODES, DHID, DHID, 0);

    // el/er
    node_dots_kernel<<<nodeWaveBlocks, 256, 0, stream>>>(z, all, arl, el, er);

    // reset per-node reductions
    fill_kernel<<<(NNODES + 255) / 256, 256, 0, stream>>>(m, NNODES, -3.0e38f);
    fill_kernel<<<(NNODES + 255) / 256, 256, 0, stream>>>(denom, NNODES, 0.0f);
    fill_kernel<<<ndBlocks, 256, 0, stream>>>(agg, (int)ND, 0.0f);

    // edge softmax + aggregation
    edge_score_max_kernel<<<edgeBlocks, 256, 0, stream>>>(src, dst, el, er, e, m);
    edge_exp_sum_kernel<<<edgeBlocks, 256, 0, stream>>>(dst, e, m, ex, denom);
    edge_aggregate_kernel<<<edgeWaveBlocks, 256, 0, stream>>>(src, dst, ex, denom, z, agg);

    // residual + bias (+ ELU except last layer)
    residual_act_kernel<<<ndBlocks, 256, 0, stream>>>(agg, bl, h, (l < NLAYER - 1) ? 1 : 0);
  }

  // readout g = segment_sum(h, graph_ids)
  fill_kernel<<<(NGRAPH * DHID + 255) / 256, 256, 0, stream>>>(g, NGRAPH * DHID, 0.0f);
  readout_kernel<<<ndBlocks, 256, 0, stream>>>(h, gid, g);

  // hidden = relu(g @ W1 + b1)
  wmma_gemm_f32<<<gemm_grid(NGRAPH, DMLP), 128, 0, stream>>>(
      g, W1, b1, hidden, NGRAPH, DHID, DMLP, 1);

  // out = hidden @ W2 + b2
  mlp_out_kernel<<<(NGRAPH * 32 + 255) / 256, 256, 0, stream>>>(hidden, W2, b2, out);
}